// InitEdge_5884105196035
// MI455X (gfx1250) — compile-verified
//
#include <hip/hip_runtime.h>
#include <hip/hip_bf16.h>
#include <math.h>

// Problem constants from the reference
#define NN   1024     // nodes
#define DN_  128      // node dim
#define DE_  128      // edge dim
#define NL   4        // layers
#define FE_  5

typedef __attribute__((ext_vector_type(2))) float v2f;
typedef __attribute__((ext_vector_type(8))) float v8f;

__device__ __forceinline__ float sigmoidf_(float x) {
    return 1.0f / (1.0f + expf(-x));
}

// ---------------------------------------------------------------------------
// Sparse adjacency-weighted edge reduction:
//   s[i][f] = sum_j adj[i][j] * e[i][j][f]
// One block per node i (128 threads, f = threadIdx.x). adj row is staged in
// LDS; rows of e are only fetched when adj[i][j] != 0 (~3% of rows), so HBM
// traffic is ~15-20 MB instead of 512 MB.
// ---------------------------------------------------------------------------
__global__ void edge_reduce_kernel(const float* __restrict__ e,
                                   const float* __restrict__ adj,
                                   float* __restrict__ s) {
    __shared__ float arow[NN];
    const int i = blockIdx.x;
    const int f = threadIdx.x;            // 0..127
    for (int j = f; j < NN; j += DE_) arow[j] = adj[i * NN + j];
    __syncthreads();
    float acc = 0.0f;
    const float* erow = e + (size_t)i * NN * DE_ + f;
    for (int j = 0; j < NN; ++j) {
        float a = arow[j];                // wave-uniform
        if (a != 0.0f) acc += a * erow[(size_t)j * DE_];
    }
    s[i * DE_ + f] = acc;
}

// ---------------------------------------------------------------------------
// Generic fp32 WMMA GEMM, one 16x16 C tile per wave (32 threads / block).
//   BT = true :  C[m][n] = sum_k A[m][k] * B[n][k]   (B is weights [N][K])
//   BT = false:  C[m][n] = sum_k A[m][k] * B[k][n]
// Optional bias[n] broadcast over rows and optional elementwise addend.
// Uses V_WMMA_F32_16X16X4_F32 (exact fp32 path).
//
// Lane layouts (ISA 7.12.2, wave32):
//   A 16x4:  lanes 0-15 -> M=lane,    K = kk+0, kk+1
//            lanes16-31 -> M=lane-16, K = kk+2, kk+3
//   B 4x16:  VGPR0 rows {kk, kk+2}, VGPR1 rows {kk+1, kk+3}, N = lane&15
//   C 16x16: VGPR v -> row m0 + v + 8*(lane>>4), col n0 + (lane&15)
// ---------------------------------------------------------------------------
template <bool BT>
__global__ __launch_bounds__(32)
void wmma_gemm_kernel(const float* __restrict__ A, int lda,
                      const float* __restrict__ B, int ldb,
                      const float* __restrict__ bias,
                      const float* __restrict__ addm,
                      float* __restrict__ C, int ldc, int K) {
    const int m0   = blockIdx.x * 16;
    const int n0   = blockIdx.y * 16;
    const int lane = threadIdx.x;         // 0..31, full wave
    const int hf   = lane >> 4;           // 0 or 1
    const int l16  = lane & 15;

    v8f acc = {};
    for (int kk = 0; kk < K; kk += 4) {
        const int ka = kk + 2 * hf;
        v2f a, b;
        a.x = A[(size_t)(m0 + l16) * lda + ka];
        a.y = A[(size_t)(m0 + l16) * lda + ka + 1];
        if (BT) {
            b.x = B[(size_t)(n0 + l16) * ldb + ka];
            b.y = B[(size_t)(n0 + l16) * ldb + ka + 1];
        } else {
            b.x = B[(size_t)ka * ldb + (n0 + l16)];
            b.y = B[(size_t)(ka + 1) * ldb + (n0 + l16)];
        }
        acc = __builtin_amdgcn_wmma_f32_16x16x4_f32(
            /*neg_a=*/false, a, /*neg_b=*/false, b,
            /*c_mod=*/(short)0, acc, /*reuse_a=*/false, /*reuse_b=*/false);
    }

    const int col = n0 + l16;
    const float bn = bias ? bias[col] : 0.0f;
#pragma unroll
    for (int v = 0; v < 8; ++v) {
        const int row = m0 + v + 8 * hf;
        float val = acc[v] + bn;
        if (addm) val += addm[(size_t)row * ldc + col];
        C[(size_t)row * ldc + col] = val;
    }
}

// ---------------------------------------------------------------------------
// GRU cell elementwise update (torch gate layout r,z,n). In-place on h.
// ---------------------------------------------------------------------------
__global__ void gru_update_kernel(const float* __restrict__ gi,
                                  const float* __restrict__ gh,
                                  float* __restrict__ h) {
    const int idx = blockIdx.x * blockDim.x + threadIdx.x;   // < N*DN
    const int i = idx >> 7;
    const int d = idx & 127;
    const float* gir = gi + (size_t)i * 3 * DN_;
    const float* ghr = gh + (size_t)i * 3 * DN_;
    const float r = sigmoidf_(gir[d] + ghr[d]);
    const float z = sigmoidf_(gir[DN_ + d] + ghr[DN_ + d]);
    const float n = tanhf(gir[2 * DN_ + d] + r * ghr[2 * DN_ + d]);
    const float hv = h[idx];
    h[idx] = (1.0f - z) * n + z * hv;
}

// hc[i] = concat(h[i], h0[i])  -> [N, 2*DN]
__global__ void concat_kernel(const float* __restrict__ h,
                              const float* __restrict__ h0,
                              float* __restrict__ hc) {
    const int idx = blockIdx.x * blockDim.x + threadIdx.x;   // < N*DN
    const int i = idx >> 7;
    const int d = idx & 127;
    hc[(size_t)i * 2 * DN_ + d]       = h[idx];
    hc[(size_t)i * 2 * DN_ + DN_ + d] = h0[idx];
}

// gv[d] = sum_i sigmoid(gpre[i][d]) * tanh(opre[i][d]),  one block of 128
__global__ void gated_reduce_kernel(const float* __restrict__ gpre,
                                    const float* __restrict__ opre,
                                    float* __restrict__ gv) {
    const int d = threadIdx.x;            // 0..127
    float acc = 0.0f;
    for (int i = 0; i < NN; ++i) {
        const float g = sigmoidf_(gpre[(size_t)i * DE_ + d]);
        const float o = tanhf(opre[(size_t)i * DE_ + d]);
        acc += g * o;
    }
    gv[d] = acc;
}

// ee = w_embed @ edge ; out = w_fc @ concat(gv, ee) + b_fc
__global__ void final_fc_kernel(const float* __restrict__ gv,
                                const float* __restrict__ w_embed,
                                const float* __restrict__ edge,
                                const float* __restrict__ w_fc,
                                const float* __restrict__ b_fc,
                                float* __restrict__ out) {
    __shared__ float ee[DE_];
    const int d = threadIdx.x;            // 0..127
    float t = 0.0f;
    for (int f = 0; f < FE_; ++f) t += w_embed[d * FE_ + f] * edge[f];
    ee[d] = t;
    __syncthreads();
    float acc = b_fc[d];
    for (int k = 0; k < DE_; ++k) acc += w_fc[d * 2 * DE_ + k] * gv[k];
    for (int k = 0; k < DE_; ++k) acc += w_fc[d * 2 * DE_ + DE_ + k] * ee[k];
    out[d] = acc;
}

// ---------------------------------------------------------------------------
// Host-side orchestration
// ---------------------------------------------------------------------------
extern "C" void kernel_launch(void* const* d_in, const int* in_sizes, int n_in,
                              void* d_out, int out_size, void* d_ws, size_t ws_size,
                              hipStream_t stream) {
    const float* h       = (const float*)d_in[0];
    const float* e       = (const float*)d_in[1];
    const float* adj     = (const float*)d_in[2];
    const float* edge    = (const float*)d_in[3];
    const float* w_msg_h = (const float*)d_in[4];
    const float* w_msg_e = (const float*)d_in[5];
    const float* b_msg   = (const float*)d_in[6];
    const float* w_ih    = (const float*)d_in[7];
    const float* w_hh    = (const float*)d_in[8];
    const float* b_ih    = (const float*)d_in[9];
    const float* b_hh    = (const float*)d_in[10];
    const float* w_gate  = (const float*)d_in[11];
    const float* b_gate  = (const float*)d_in[12];
    const float* w_out   = (const float*)d_in[13];
    const float* b_out   = (const float*)d_in[14];
    const float* w_embed = (const float*)d_in[15];
    const float* w_fc    = (const float*)d_in[16];
    const float* b_fc    = (const float*)d_in[17];
    float* out = (float*)d_out;

    // Workspace layout (floats); ~7.9 MB total.
    float* ws   = (float*)d_ws;
    float* s    = ws;                        // [N, DE]    1024*128
    float* me   = s    + NN * DE_;           // [N, DN]
    float* hcur = me   + NN * DN_;           // [N, DN]
    float* hW   = hcur + NN * DN_;           // [N, DN]
    float* m    = hW   + NN * DN_;           // [N, DN]
    float* gi   = m    + NN * DN_;           // [N, 3*DN]
    float* gh   = gi   + NN * 3 * DN_;       // [N, 3*DN]
    float* hc   = gh   + NN * 3 * DN_;       // [N, 2*DN]
    float* gpre = hc   + NN * 2 * DN_;       // [N, DE]
    float* opre = gpre + NN * DE_;           // [N, DE]
    float* gv   = opre + NN * DE_;           // [DE]

    // h_cur = h
    hipMemcpyAsync(hcur, h, (size_t)NN * DN_ * sizeof(float),
                   hipMemcpyDeviceToDevice, stream);

    // s[i][f] = sum_j adj[i][j]*e[i][j][f]   (layer-invariant, reads e once)
    edge_reduce_kernel<<<NN, DE_, 0, stream>>>(e, adj, s);

    // m_e = s @ w_msg_e^T + b_msg            (layer-invariant)
    wmma_gemm_kernel<true><<<dim3(NN / 16, DN_ / 16), 32, 0, stream>>>(
        s, DE_, w_msg_e, DE_, b_msg, nullptr, me, DN_, DE_);

    const int ew_blocks = (NN * DN_) / 256;
    for (int l = 0; l < NL; ++l) {
        // hW = h @ w_msg_h^T
        wmma_gemm_kernel<true><<<dim3(NN / 16, DN_ / 16), 32, 0, stream>>>(
            hcur, DN_, w_msg_h, DN_, nullptr, nullptr, hW, DN_, DN_);
        // m = adj @ hW + m_e
        wmma_gemm_kernel<false><<<dim3(NN / 16, DN_ / 16), 32, 0, stream>>>(
            adj, NN, hW, DN_, nullptr, me, m, DN_, NN);
        // gi = m @ w_ih^T + b_ih   [N, 3*DN]
        wmma_gemm_kernel<true><<<dim3(NN / 16, (3 * DN_) / 16), 32, 0, stream>>>(
            m, DN_, w_ih, DN_, b_ih, nullptr, gi, 3 * DN_, DN_);
        // gh = h @ w_hh^T + b_hh   [N, 3*DN]
        wmma_gemm_kernel<true><<<dim3(NN / 16, (3 * DN_) / 16), 32, 0, stream>>>(
            hcur, DN_, w_hh, DN_, b_hh, nullptr, gh, 3 * DN_, DN_);
        // GRU elementwise (in-place on hcur)
        gru_update_kernel<<<ew_blocks, 256, 0, stream>>>(gi, gh, hcur);
    }

    // Readout
    concat_kernel<<<ew_blocks, 256, 0, stream>>>(hcur, h, hc);
    wmma_gemm_kernel<true><<<dim3(NN / 16, DE_ / 16), 32, 0, stream>>>(
        hc, 2 * DN_, w_gate, 2 * DN_, b_gate, nullptr, gpre, DE_, 2 * DN_);
    wmma_gemm_kernel<true><<<dim3(NN / 16, DE_ / 16), 32, 0, stream>>>(
        hc, 2 * DN_, w_out, 2 * DN_, b_out, nullptr, opre, DE_, 2 * DN_);
    gated_reduce_kernel<<<1, DE_, 0, stream>>>(gpre, opre, gv);
    final_fc_kernel<<<1, DE_, 0, stream>>>(gv, w_embed, edge, w_fc, b_fc, out);
}